// GINEncoder_19636590478048
// MI455X (gfx1250) — compile-verified
//
#include <hip/hip_runtime.h>
#include <math.h>

// ---------------------------------------------------------------------------
// GIN encoder, MI455X (gfx1250, wave32).
// N=100000 nodes, D=128, E=1.6M edges, L=3 layers.
// GEMMs on V_WMMA_F32_16X16X4_F32 (fp32 matrix pipe, exact accuracy).
// Edge scatter uses native GLOBAL_ATOMIC_ADD_F32 (no-return, STOREcnt).
// ---------------------------------------------------------------------------

#define DF 128                 // feature dim
#define AS_STRIDE 132          // A tile LDS stride (floats): conflict-free frag reads
#define BS_STRIDE 288          // B tile LDS stride (floats) = 144 float2 pair-rows

typedef float v2f __attribute__((ext_vector_type(2)));
typedef float v8f __attribute__((ext_vector_type(8)));

__device__ __forceinline__ v8f wmma_f32(v2f a, v2f b, v8f c) {
  // 8 args: (neg_a, A, neg_b, B, c_mod, C, reuse_a, reuse_b)
  return __builtin_amdgcn_wmma_f32_16x16x4_f32(false, a, false, b, (short)0, c,
                                               false, false);
}

__device__ __forceinline__ float gelu_exact(float x) {
  return 0.5f * x * (1.0f + erff(x * 0.70710678118654752f));
}

// Native fire-and-forget f32 atomic add (global_atomic_add_f32, no CAS loop).
__device__ __forceinline__ void atomic_add_f32(float* p, float v) {
  unsafeAtomicAdd(p, v);
}

// --------------------------- elementwise kernels ---------------------------

__global__ void k_init(const float* __restrict__ x, float* __restrict__ h,
                       float* __restrict__ hsum, int nvec4) {
  int i = blockIdx.x * blockDim.x + threadIdx.x;
  if (i < nvec4) {
    float4 v = ((const float4*)x)[i];
    ((float4*)h)[i] = v;
    ((float4*)hsum)[i] = make_float4(0.f, 0.f, 0.f, 0.f);
  }
}

__global__ void k_prescale(const float* __restrict__ h, float* __restrict__ z,
                           const float* __restrict__ eps, int layer, int nvec4) {
  int i = blockIdx.x * blockDim.x + threadIdx.x;
  if (i < nvec4) {
    float s = 1.0f + eps[layer];
    float4 v = ((const float4*)h)[i];
    v.x *= s; v.y *= s; v.z *= s; v.w *= s;
    ((float4*)z)[i] = v;
  }
}

// One edge handled by 32 consecutive threads (float4 per lane): gather h[src],
// atomic-add into z[dst]. h and z are L2-resident (51.2 MB each, 192 MB L2),
// so this phase is bounded by L2 atomic throughput -> must be native fadd.
// Block = 256 threads = exactly 8 edges; row offsets computed in 64-bit (size_t).
__global__ void k_scatter(const float* __restrict__ h, float* __restrict__ z,
                          const int* __restrict__ src, const int* __restrict__ dst,
                          int nE) {
  int e = blockIdx.x * 8 + (threadIdx.x >> 5);
  if (e >= nE) return;
  // Stream-prefetch indices a few blocks ahead (global_prefetch_b8).
  if (threadIdx.x == 0 && e + 64 < nE) {
    __builtin_prefetch(src + e + 64, 0, 1);
    __builtin_prefetch(dst + e + 64, 0, 1);
  }
  int c = (threadIdx.x & 31) << 2;
  int s = src[e], d = dst[e];
  const float4 v = *(const float4*)(h + (size_t)s * DF + c);
  float* p = z + (size_t)d * DF + c;
  atomic_add_f32(p + 0, v.x);
  atomic_add_f32(p + 1, v.y);
  atomic_add_f32(p + 2, v.z);
  atomic_add_f32(p + 3, v.w);
}

// Final pre-projection LayerNorm: one wave per row (32 lanes x float4).
__global__ void k_layernorm(const float* __restrict__ X, const float* __restrict__ g,
                            const float* __restrict__ b, float* __restrict__ Z,
                            int Mrows) {
  int wave = threadIdx.x >> 5, lane = threadIdx.x & 31;
  int r = blockIdx.x * 8 + wave;
  if (r >= Mrows) return;
  const float4 v = *(const float4*)(X + (size_t)r * DF + lane * 4);
  float s = v.x + v.y + v.z + v.w;
  float sq = v.x * v.x + v.y * v.y + v.z * v.z + v.w * v.w;
#pragma unroll
  for (int o = 1; o < 32; o <<= 1) {
    s += __shfl_xor(s, o);
    sq += __shfl_xor(sq, o);
  }
  float mean = s * (1.0f / 128.0f);
  float var = sq * (1.0f / 128.0f) - mean * mean;
  float rs = rsqrtf(var + 1e-5f);
  float4 gg = *(const float4*)(g + lane * 4);
  float4 bb = *(const float4*)(b + lane * 4);
  float4 o;
  o.x = (v.x - mean) * rs * gg.x + bb.x;
  o.y = (v.y - mean) * rs * gg.y + bb.y;
  o.z = (v.z - mean) * rs * gg.z + bb.z;
  o.w = (v.w - mean) * rs * gg.w + bb.w;
  *(float4*)(Z + (size_t)r * DF + lane * 4) = o;
}

// ------------------------------ GEMM helpers -------------------------------
// Block: 256 threads (8 waves). Tile: 64 rows x 128 cols, K=128 fully in LDS.
// A LDS: row-major, stride 132.  W LDS: K-pair packed, pair-row stride 144
// float2 so a B fragment (b.x=B[k+2kh][n], b.y=B[k+2kh+1][n]) is 1 ds_load_b64
// and the two half-wave lane groups hit disjoint LDS bank halves.

__device__ __forceinline__ void load_tiles(const float* __restrict__ A,
                                           const float* __restrict__ W,
                                           float* As, float* Bs, int m0, int Mrows) {
  int tid = threadIdx.x;
  // A tile: 64x128 floats = 2048 float4, 8 per thread.
#pragma unroll
  for (int j = 0; j < 8; j++) {
    int f = tid + j * 256;
    int r = f >> 5, c = (f & 31) << 2;
    int g = m0 + r;
    float4 v = make_float4(0.f, 0.f, 0.f, 0.f);
    if (g < Mrows) v = *(const float4*)(A + (size_t)g * DF + c);
    *(float4*)(As + r * AS_STRIDE + c) = v;
  }
  // W: 128x128 floats = 4096 float4, 16 per thread, scattered into pair rows.
#pragma unroll
  for (int j = 0; j < 16; j++) {
    int f = tid + j * 256;
    int r = f >> 5, c = (f & 31) << 2;
    float4 v = *(const float4*)(W + r * DF + c);
    float* bp = Bs + (r >> 1) * BS_STRIDE + (r & 1);
    bp[(c + 0) * 2] = v.x;
    bp[(c + 1) * 2] = v.y;
    bp[(c + 2) * 2] = v.z;
    bp[(c + 3) * 2] = v.w;
  }
}

// Each wave: rows rb*16..+15, cols cb*64..+63 (4 tiles of 16), A frag reused 4x.
__device__ __forceinline__ void mma_64x128(const float* As, const float* Bs,
                                           v8f acc[4]) {
  int lane = threadIdx.x & 31;
  int wave = threadIdx.x >> 5;
  int rb = wave >> 1, cb = wave & 1;
  int row16 = lane & 15, kh = lane >> 4;
  const float* arow = As + (rb * 16 + row16) * AS_STRIDE + 2 * kh;
  for (int k = 0; k < DF; k += 4) {
    v2f a = *(const v2f*)(arow + k);
    const float* brow = Bs + ((k >> 1) + kh) * BS_STRIDE;
#pragma unroll
    for (int t = 0; t < 4; t++) {
      int col = cb * 64 + t * 16 + row16;
      v2f b = *(const v2f*)(brow + col * 2);
      acc[t] = wmma_f32(a, b, acc[t]);
    }
  }
}

// GEMM with bias (+ optional ReLU) epilogue: y = act(A @ W + bias).
template <bool RELU>
__global__ void k_gemm_bias_act(const float* __restrict__ A, const float* __restrict__ W,
                                const float* __restrict__ bias, float* __restrict__ Y,
                                int Mrows) {
  __shared__ float As[64 * AS_STRIDE];
  __shared__ float Bs[64 * BS_STRIDE];
  int m0 = blockIdx.x * 64;
  load_tiles(A, W, As, Bs, m0, Mrows);
  __syncthreads();
  v8f zero = {0.f, 0.f, 0.f, 0.f, 0.f, 0.f, 0.f, 0.f};
  v8f acc[4] = {zero, zero, zero, zero};
  mma_64x128(As, Bs, acc);
  int lane = threadIdx.x & 31, wave = threadIdx.x >> 5;
  int rb = wave >> 1, cb = wave & 1, row16 = lane & 15, kh = lane >> 4;
#pragma unroll
  for (int t = 0; t < 4; t++) {
    int col = cb * 64 + t * 16 + row16;
    float bb = bias[col];
#pragma unroll
    for (int j = 0; j < 8; j++) {
      int r = m0 + rb * 16 + kh * 8 + j;
      if (r < Mrows) {
        float v = acc[t][j] + bb;
        if (RELU) v = fmaxf(v, 0.f);
        Y[(size_t)r * DF + col] = v;
      }
    }
  }
}

// GEMM2 + fused per-row LayerNorm + exact GELU + residual + h_sum accumulate.
__global__ void k_gemm_ln_gelu_res(const float* __restrict__ A, const float* __restrict__ W,
                                   const float* __restrict__ bias,
                                   const float* __restrict__ g, const float* __restrict__ bln,
                                   float* __restrict__ h, float* __restrict__ hsum,
                                   int Mrows) {
  __shared__ float As[64 * AS_STRIDE];
  __shared__ float Bs[64 * BS_STRIDE];
  int m0 = blockIdx.x * 64;
  load_tiles(A, W, As, Bs, m0, Mrows);
  __syncthreads();
  v8f zero = {0.f, 0.f, 0.f, 0.f, 0.f, 0.f, 0.f, 0.f};
  v8f acc[4] = {zero, zero, zero, zero};
  mma_64x128(As, Bs, acc);
  __syncthreads();  // all As reads complete; reuse As for the C tile
  int lane = threadIdx.x & 31, wave = threadIdx.x >> 5;
  int rb = wave >> 1, cb = wave & 1, row16 = lane & 15, kh = lane >> 4;
#pragma unroll
  for (int t = 0; t < 4; t++) {
    int col = cb * 64 + t * 16 + row16;
    float bb = bias[col];
#pragma unroll
    for (int j = 0; j < 8; j++)
      As[(rb * 16 + kh * 8 + j) * AS_STRIDE + col] = acc[t][j] + bb;
  }
  __syncthreads();
  // LayerNorm over 128 cols: 4 threads per row (quad), 32 cols each.
  int r = threadIdx.x >> 2;
  int q = threadIdx.x & 3;
  int grow = m0 + r;
  const float* crow = As + r * AS_STRIDE + q * 32;
  float s = 0.f, sq = 0.f;
#pragma unroll
  for (int c = 0; c < 32; c++) {
    float x = crow[c];
    s += x;
    sq += x * x;
  }
  s += __shfl_xor(s, 1);  s += __shfl_xor(s, 2);
  sq += __shfl_xor(sq, 1); sq += __shfl_xor(sq, 2);
  float mean = s * (1.0f / 128.0f);
  float var = sq * (1.0f / 128.0f) - mean * mean;
  float rs = rsqrtf(var + 1e-5f);
  if (grow < Mrows) {
    float* hp = h + (size_t)grow * DF + q * 32;
    float* sp = hsum + (size_t)grow * DF + q * 32;
#pragma unroll
    for (int c = 0; c < 32; c++) {
      float x = (crow[c] - mean) * rs * g[q * 32 + c] + bln[q * 32 + c];
      float hn = gelu_exact(x) + hp[c];
      hp[c] = hn;
      sp[c] += hn;
    }
  }
}

// ------------------------------- launcher ----------------------------------

extern "C" void kernel_launch(void* const* d_in, const int* in_sizes, int n_in,
                              void* d_out, int out_size, void* d_ws, size_t ws_size,
                              hipStream_t stream) {
  const float* x     = (const float*)d_in[0];
  const int*   ei    = (const int*)d_in[1];
  const float* W1    = (const float*)d_in[2];
  const float* b1    = (const float*)d_in[3];
  const float* W2    = (const float*)d_in[4];
  const float* b2    = (const float*)d_in[5];
  const float* eps   = (const float*)d_in[6];
  const float* ln_g  = (const float*)d_in[7];
  const float* ln_b  = (const float*)d_in[8];
  const float* pln_g = (const float*)d_in[9];
  const float* pln_b = (const float*)d_in[10];
  const float* pW    = (const float*)d_in[11];
  const float* pb    = (const float*)d_in[12];
  float* out = (float*)d_out;

  const int N = in_sizes[0] / DF;
  const int E = in_sizes[1] / 2;
  const int L = in_sizes[6];

  float* h    = (float*)d_ws;
  float* hsum = h + (size_t)N * DF;
  float* z    = hsum + (size_t)N * DF;
  float* y1   = z + (size_t)N * DF;

  const int nvec4 = N * DF / 4;
  k_init<<<(nvec4 + 255) / 256, 256, 0, stream>>>(x, h, hsum, nvec4);

  dim3 ggrid((N + 63) / 64);
  unsigned sgrid = (unsigned)((E + 7) / 8);  // 8 edges per 256-thread block

  for (int l = 0; l < L; l++) {
    k_prescale<<<(nvec4 + 255) / 256, 256, 0, stream>>>(h, z, eps, l, nvec4);
    k_scatter<<<sgrid, 256, 0, stream>>>(h, z, ei, ei + E, E);
    k_gemm_bias_act<true><<<ggrid, 256, 0, stream>>>(
        z, W1 + (size_t)l * DF * DF, b1 + (size_t)l * DF, y1, N);
    k_gemm_ln_gelu_res<<<ggrid, 256, 0, stream>>>(
        y1, W2 + (size_t)l * DF * DF, b2 + (size_t)l * DF,
        ln_g + (size_t)l * DF, ln_b + (size_t)l * DF, h, hsum, N);
  }
  k_layernorm<<<(N + 7) / 8, 256, 0, stream>>>(hsum, pln_g, pln_b, z, N);
  k_gemm_bias_act<false><<<ggrid, 256, 0, stream>>>(z, pW, pb, out, N);
}